// DGCNNEncoder_26431228739761
// MI455X (gfx1250) — compile-verified
//
#include <hip/hip_runtime.h>
#include <math.h>

// ---------------------------------------------------------------------------
// DGCNN encoder for MI455X (gfx1250, wave32).
// All GEMM-shaped work (KNN distance matrix, edge-conv, final MLP) runs on
// V_WMMA_F32_16X16X4_F32 (fp32-exact). Operand staging uses the CDNA5 async
// global->LDS data movers where available.
// ---------------------------------------------------------------------------

typedef __attribute__((ext_vector_type(2))) float v2f;
typedef __attribute__((ext_vector_type(8))) float v8f;

#define NPTS 2048
#define KNN  20
#define NEG_SLOPE 0.2f
#define BN_EPS 1e-5f
#define FLT_BIG 3.402823e38f

// D = A(16x4) * B(4x16) + C, fp32.
// A frag: lane l holds A[l%16][c0], A[l%16][c0+1], c0 = 2*(l/16)
// B frag: lane l holds B[k0][l%16], B[k0+1][l%16],  k0 = 2*(l/16)
// C/D:    lane l holds rows (8*(l/16) + 0..7) of column l%16
__device__ __forceinline__ v8f wmma4(v2f a, v2f b, v8f c) {
  return __builtin_amdgcn_wmma_f32_16x16x4_f32(
      false, a, false, b, (short)0, c, false, false);
}

// ---- CDNA5 async global -> LDS copy (ASYNCcnt-tracked), with fallback ------
#if __has_builtin(__builtin_amdgcn_global_load_async_to_lds_b128)
#define HAS_ASYNC_LDS 1
#else
#define HAS_ASYNC_LDS 0
#endif

__device__ __forceinline__ void cp_async_b128(float* lds, const float* g) {
#if HAS_ASYNC_LDS
  typedef int vi4 __attribute__((vector_size(4 * sizeof(int))));
  __builtin_amdgcn_global_load_async_to_lds_b128(
      (__attribute__((address_space(1))) vi4*)(unsigned long long)(const void*)g,
      (__attribute__((address_space(3))) vi4*)(unsigned int)(unsigned long long)(void*)lds,
      0, 0);
#else
  *(float4*)lds = *(const float4*)g;
#endif
}

__device__ __forceinline__ void cp_async_wait() {
#if HAS_ASYNC_LDS && __has_builtin(__builtin_amdgcn_s_wait_asynccnt)
  __builtin_amdgcn_s_wait_asynccnt(0);
#endif
}

// ---------------------------------------------------------------------------
// 1) per-point squared norms
// ---------------------------------------------------------------------------
__global__ void sq_kernel(const float* __restrict__ x, long long bstride,
                          int ld, int C, float* __restrict__ sq) {
  const int b = blockIdx.y;
  const int n = blockIdx.x * blockDim.x + threadIdx.x;
  if (n >= NPTS) return;
  const float* p = x + (long long)b * bstride + (long long)n * ld;
  float s = 0.f;
  for (int c = 0; c < C; ++c) s += p[c] * p[c];
  sq[b * NPTS + n] = s;
}

// ---------------------------------------------------------------------------
// 2) KNN: block = 16 query points, 8 waves. WMMA fp32 distance tiles into a
//    16 x 2048 LDS row-block (128 KB of the WGP's 320 KB), then top-20
//    selection with 16 threads per query row. Query fragments are hoisted
//    into registers (reused across all 128 m-tiles).
// ---------------------------------------------------------------------------
template <int C>
__global__ __launch_bounds__(256) void knn_kernel(
    const float* __restrict__ x, long long bstride, int ld,
    const float* __restrict__ sq, int* __restrict__ knn_idx) {
  constexpr int C4 = (C + 3) / 4;
  __shared__ float dist[16 * NPTS];
  __shared__ float sqq[16];
  __shared__ float redD[256];
  __shared__ int   redI[256];

  const int b = blockIdx.y;
  const int qbase = blockIdx.x * 16;
  const int tid  = threadIdx.x;
  const int wave = tid >> 5;
  const int lane = tid & 31;
  const int lh = lane >> 4;
  const int ll = lane & 15;

  if (tid < 16) sqq[tid] = sq[b * NPTS + qbase + tid];
  __syncthreads();

  const float* xb = x + (long long)b * bstride;
  const int qrow = qbase + ll;

  // hoisted query-side A fragments
  v2f af[C4];
#pragma unroll
  for (int kc = 0; kc < C4; ++kc) {
    const int c0 = kc * 4 + lh * 2;
    if constexpr (C % 4 == 0) {
      af[kc] = *(const v2f*)(xb + (long long)qrow * ld + c0);
    } else {
      af[kc].x = (c0     < C) ? xb[(long long)qrow * ld + c0]     : 0.f;
      af[kc].y = (c0 + 1 < C) ? xb[(long long)qrow * ld + c0 + 1] : 0.f;
    }
  }

  for (int mt = wave; mt < NPTS / 16; mt += 8) {
    const int mbase = mt * 16;
    const int mrow = mbase + ll;
    const float* xm = xb + (long long)mrow * ld;
    v8f acc = {};
#pragma unroll
    for (int kc = 0; kc < C4; ++kc) {
      const int c0 = kc * 4 + lh * 2;
      v2f bb;
      if constexpr (C % 4 == 0) {
        bb = *(const v2f*)(xm + c0);
      } else {
        bb.x = (c0     < C) ? xm[c0]     : 0.f;
        bb.y = (c0 + 1 < C) ? xm[c0 + 1] : 0.f;
      }
      acc = wmma4(af[kc], bb, acc);
    }
    const float sqm = sq[b * NPTS + mrow];
#pragma unroll
    for (int r = 0; r < 8; ++r) {
      const int qloc = lh * 8 + r;
      dist[qloc * NPTS + mbase + ll] = sqq[qloc] + sqm - 2.f * acc[r];
    }
  }
  __syncthreads();

  // top-K: group g (16 threads) owns query row g
  const int g = tid >> 4;
  const int t = tid & 15;
  for (int sel = 0; sel < KNN; ++sel) {
    float best = FLT_BIG; int bi = 0x7fffffff;
    for (int m = t; m < NPTS; m += 16) {
      const float v = dist[g * NPTS + m];
      if (v < best) { best = v; bi = m; }
    }
    redD[tid] = best; redI[tid] = bi;
    __syncthreads();
    if (t == 0) {
      float bv = redD[g * 16]; int bix = redI[g * 16];
      for (int u = 1; u < 16; ++u) {
        const float v = redD[g * 16 + u]; const int i2 = redI[g * 16 + u];
        if (v < bv || (v == bv && i2 < bix)) { bv = v; bix = i2; }
      }
      knn_idx[((long long)b * NPTS + qbase + g) * KNN + sel] = bix;
      dist[g * NPTS + bix] = FLT_BIG;
    }
    __syncthreads();
  }
}

// ---------------------------------------------------------------------------
// 3) Edge conv: block owns PPB points; neighbor + center features staged to
//    LDS once (async b128 where possible), then each wave computes 16x16
//    output tiles with WMMA. BN + LeakyReLU + max-over-k fused in-register.
// ---------------------------------------------------------------------------
template <int C, int O, int PPB>
__global__ __launch_bounds__(256) void edgeconv_kernel(
    const float* __restrict__ xin, long long xbstride, int xld,
    const int* __restrict__ knn_idx,
    const float* __restrict__ W, const float* __restrict__ bn,
    float* __restrict__ out, long long obstride, int oldd, int ooff) {
  constexpr int C2 = 2 * C;
  constexpr int C4 = (C2 + 3) / 4;
  constexpr int OT = O / 16;
  constexpr int JOBS = PPB * OT;          // multiple of 8 by construction
  constexpr int CROW = (C % 4 == 0) ? C : (((C + 3) / 4) * 4);

  __shared__ __attribute__((aligned(16))) float nbrF[PPB][32][CROW];
  __shared__ __attribute__((aligned(16))) float ctrF[PPB][CROW];
  __shared__ int nbrI[PPB][32];

  const int tid = threadIdx.x;
  const int b = blockIdx.y;
  const int n0 = blockIdx.x * PPB;

  // neighbor indices (pad k>=20 with self; masked in epilogue)
  for (int i = tid; i < PPB * 32; i += 256) {
    const int p = i >> 5, k = i & 31;
    nbrI[p][k] = (k < KNN)
        ? knn_idx[((long long)b * NPTS + n0 + p) * KNN + k] : (n0 + p);
  }
  __syncthreads();

  const float* xb = xin + (long long)b * xbstride;
  if constexpr (C % 4 == 0) {
    constexpr int CW = C / 4;             // 16-byte words per row
    for (int i = tid; i < PPB * 32 * CW; i += 256) {
      const int p = i / (32 * CW);
      const int r = i % (32 * CW);
      const int k = r / CW, c4 = r % CW;
      cp_async_b128(&nbrF[p][k][c4 * 4],
                    xb + (long long)nbrI[p][k] * xld + c4 * 4);
    }
    for (int i = tid; i < PPB * CW; i += 256) {
      const int p = i / CW, c4 = i % CW;
      cp_async_b128(&ctrF[p][c4 * 4], xb + (long long)(n0 + p) * xld + c4 * 4);
    }
    cp_async_wait();
  } else {
    for (int i = tid; i < PPB * 32 * C; i += 256) {
      const int p = i / (32 * C);
      const int r = i % (32 * C);
      const int k = r / C, c = r % C;
      nbrF[p][k][c] = xb[(long long)nbrI[p][k] * xld + c];
    }
    for (int i = tid; i < PPB * C; i += 256) {
      const int p = i / C, c = i % C;
      ctrF[p][c] = xb[(long long)(n0 + p) * xld + c];
    }
  }
  __syncthreads();

  const int lane = tid & 31, wave = tid >> 5;
  const int lh = lane >> 4, ll = lane & 15;

  auto eAt = [&](int p, int k, int c) -> float {   // scalar path (C=3)
    if (c >= C2) return 0.f;
    if (c < C) return nbrF[p][k][c] - ctrF[p][c];
    return ctrF[p][c - C];
  };

  for (int job = wave; job < JOBS; job += 8) {
    const int p = job / OT, ot = job % OT;
    const int n = n0 + p;
    const int o = ot * 16 + ll;
    const float* wrow = W + (long long)o * C2;

    v8f acc0 = {}, acc1 = {};
#pragma unroll
    for (int kc = 0; kc < C4; ++kc) {
      const int c0 = kc * 4 + lh * 2;
      v2f a0, a1, bb;
      if constexpr (C % 4 == 0) {
        if (kc * 4 < C) {                 // folds after unroll: (nbr - ctr)
          const v2f ctr = *(const v2f*)&ctrF[p][c0];
          a0 = *(const v2f*)&nbrF[p][ll][c0] - ctr;
          a1 = *(const v2f*)&nbrF[p][16 + ll][c0] - ctr;
        } else {                          // center copy half
          const v2f ctr = *(const v2f*)&ctrF[p][c0 - C];
          a0 = ctr; a1 = ctr;
        }
      } else {
        a0.x = eAt(p, ll, c0);      a0.y = eAt(p, ll, c0 + 1);
        a1.x = eAt(p, 16 + ll, c0); a1.y = eAt(p, 16 + ll, c0 + 1);
      }
      if constexpr (C2 % 4 == 0) {
        bb = *(const v2f*)(wrow + c0);
      } else {
        bb.x = (c0     < C2) ? wrow[c0]     : 0.f;
        bb.y = (c0 + 1 < C2) ? wrow[c0 + 1] : 0.f;
      }
      acc0 = wmma4(a0, bb, acc0);
      acc1 = wmma4(a1, bb, acc1);
    }

    const float gam  = bn[o];
    const float beta = bn[O + o];
    const float mean = bn[2 * O + o];
    const float var  = bn[3 * O + o];
    const float scale = gam * rsqrtf(var + BN_EPS);

    float mx = -FLT_BIG;
#pragma unroll
    for (int r = 0; r < 8; ++r) {
      float y0 = (acc0[r] - mean) * scale + beta;       // k = lh*8+r < 16
      y0 = (y0 > 0.f) ? y0 : NEG_SLOPE * y0;
      mx = fmaxf(mx, y0);
      const int k1 = 16 + lh * 8 + r;
      float y1 = (acc1[r] - mean) * scale + beta;
      y1 = (y1 > 0.f) ? y1 : NEG_SLOPE * y1;
      if (k1 < KNN) mx = fmaxf(mx, y1);                 // mask pad rows
    }
    const float other = __shfl_xor(mx, 16, 32);
    mx = fmaxf(mx, other);
    if (lane < 16)
      out[(long long)b * obstride + (long long)n * oldd + ooff + o] = mx;
  }
}

// ---------------------------------------------------------------------------
// 4) Final MLP: block owns one 16-row tile of xcat, staged once into LDS
//    (async), reused across all 64 output tiles. feat = lrelu(bn5(x @ W5^T)).
// ---------------------------------------------------------------------------
__global__ __launch_bounds__(256) void mlp_kernel(
    const float* __restrict__ xcat, const float* __restrict__ W5,
    const float* __restrict__ bn5, float* __restrict__ feat) {
  __shared__ __attribute__((aligned(16))) float arow[16][512];   // 32 KB

  const int mt = blockIdx.x;              // 0 .. B*N/16-1
  const int tid = threadIdx.x;
  {
    const float* src = xcat + (long long)mt * 16 * 512;
    for (int i = tid; i < 16 * 512 / 4; i += 256)
      cp_async_b128(&arow[0][0] + i * 4, src + i * 4);
    cp_async_wait();
  }
  __syncthreads();

  const int lane = tid & 31, wave = tid >> 5;
  const int lh = lane >> 4, ll = lane & 15;

  for (int ot = wave; ot < 64; ot += 8) {
    const int o = ot * 16 + ll;
    const float* wrow = W5 + (long long)o * 512;
    v8f acc = {};
#pragma unroll 16
    for (int kc = 0; kc < 128; ++kc) {
      const int c0 = kc * 4 + lh * 2;
      const v2f a  = *(const v2f*)&arow[ll][c0];
      const v2f bb = *(const v2f*)(wrow + c0);
      acc = wmma4(a, bb, acc);
    }
    const float gam  = bn5[o];
    const float beta = bn5[1024 + o];
    const float mean = bn5[2048 + o];
    const float var  = bn5[3072 + o];
    const float scale = gam * rsqrtf(var + BN_EPS);
#pragma unroll
    for (int r = 0; r < 8; ++r) {
      const int rr = mt * 16 + lh * 8 + r;
      float y = (acc[r] - mean) * scale + beta;
      y = (y > 0.f) ? y : NEG_SLOPE * y;
      feat[(long long)rr * 1024 + o] = y;
    }
  }
}

// ---------------------------------------------------------------------------
// 5) Global max + mean pooling over N, concatenated output (B, 2048)
// ---------------------------------------------------------------------------
__global__ void pool_kernel(const float* __restrict__ feat,
                            float* __restrict__ out) {
  const int b = blockIdx.y;
  const int o = blockIdx.x * blockDim.x + threadIdx.x;   // 0..1023
  const float* p = feat + (long long)b * NPTS * 1024 + o;
  float mx = -FLT_BIG, sum = 0.f;
  for (int n = 0; n < NPTS; ++n) {
    if (n + 16 < NPTS) __builtin_prefetch(&p[(long long)(n + 16) * 1024], 0, 1);
    const float v = p[(long long)n * 1024];
    mx = fmaxf(mx, v);
    sum += v;
  }
  out[(long long)b * 2048 + o]        = mx;
  out[(long long)b * 2048 + 1024 + o] = sum * (1.f / NPTS);
}

// ---------------------------------------------------------------------------
// Orchestration
// ---------------------------------------------------------------------------
extern "C" void kernel_launch(void* const* d_in, const int* in_sizes, int n_in,
                              void* d_out, int out_size, void* d_ws,
                              size_t ws_size, hipStream_t stream) {
  const float* x   = (const float*)d_in[0];
  const float* W1  = (const float*)d_in[1];
  const float* W2  = (const float*)d_in[2];
  const float* W3  = (const float*)d_in[3];
  const float* W4  = (const float*)d_in[4];
  const float* W5  = (const float*)d_in[5];
  const float* bn1 = (const float*)d_in[6];
  const float* bn2 = (const float*)d_in[7];
  const float* bn3 = (const float*)d_in[8];
  const float* bn4 = (const float*)d_in[9];
  const float* bn5 = (const float*)d_in[10];

  const int B = 8;
  char* ws = (char*)d_ws;
  float* sq   = (float*)ws;                                  //  64 KB
  int*   idx  = (int*)(ws + (size_t)B * NPTS * 4);           // 1.25 MB
  float* xcat = (float*)(ws + (size_t)B * NPTS * 4
                            + (size_t)B * NPTS * KNN * 4);   // (B,N,512) 33.5 MB
  float* feat = (float*)((char*)xcat + (size_t)B * NPTS * 512 * 4); // 67 MB

  const long long xbs = (long long)NPTS * 3;
  const long long cbs = (long long)NPTS * 512;

  dim3 blk(256);
  dim3 gSq(NPTS / 256, B);
  dim3 gKnn(NPTS / 16, B);

  // ---- layer 1: C=3 -> O=64, output -> xcat[:, 0:64)
  sq_kernel<<<gSq, blk, 0, stream>>>(x, xbs, 3, 3, sq);
  knn_kernel<3><<<gKnn, blk, 0, stream>>>(x, xbs, 3, sq, idx);
  edgeconv_kernel<3, 64, 2><<<dim3(NPTS / 2, B), blk, 0, stream>>>(
      x, xbs, 3, idx, W1, bn1, xcat, cbs, 512, 0);

  // ---- layer 2: C=64 -> O=64, input xcat[:,0:64), output xcat[:,64:128)
  sq_kernel<<<gSq, blk, 0, stream>>>(xcat + 0, cbs, 512, 64, sq);
  knn_kernel<64><<<gKnn, blk, 0, stream>>>(xcat + 0, cbs, 512, sq, idx);
  edgeconv_kernel<64, 64, 2><<<dim3(NPTS / 2, B), blk, 0, stream>>>(
      xcat + 0, cbs, 512, idx, W2, bn2, xcat, cbs, 512, 64);

  // ---- layer 3: C=64 -> O=128, input xcat[:,64:128), output xcat[:,128:256)
  sq_kernel<<<gSq, blk, 0, stream>>>(xcat + 64, cbs, 512, 64, sq);
  knn_kernel<64><<<gKnn, blk, 0, stream>>>(xcat + 64, cbs, 512, sq, idx);
  edgeconv_kernel<64, 128, 1><<<dim3(NPTS, B), blk, 0, stream>>>(
      xcat + 64, cbs, 512, idx, W3, bn3, xcat, cbs, 512, 128);

  // ---- layer 4: C=128 -> O=256, input xcat[:,128:256), output xcat[:,256:512)
  sq_kernel<<<gSq, blk, 0, stream>>>(xcat + 128, cbs, 512, 128, sq);
  knn_kernel<128><<<gKnn, blk, 0, stream>>>(xcat + 128, cbs, 512, sq, idx);
  edgeconv_kernel<128, 256, 1><<<dim3(NPTS, B), blk, 0, stream>>>(
      xcat + 128, cbs, 512, idx, W4, bn4, xcat, cbs, 512, 256);

  // ---- final MLP (B*N,512) x (512,1024)^T + BN + LReLU
  mlp_kernel<<<dim3(B * NPTS / 16), blk, 0, stream>>>(xcat, W5, bn5, feat);

  // ---- global max/mean pool -> (B, 2048)
  pool_kernel<<<dim3(1024 / 256, B), blk, 0, stream>>>(feat, (float*)d_out);
}